// L0Contraction_20650202759259
// MI455X (gfx1250) — compile-verified
//
#include <hip/hip_runtime.h>

typedef __attribute__((ext_vector_type(2))) float v2f;
typedef __attribute__((ext_vector_type(4))) float v4f;
typedef __attribute__((ext_vector_type(8))) float v8f;

#define WAVES_PER_BLOCK 8
#define ROWS_PER_TILE   128
#define MTOT            15
#define LDS_STRIDE      17                      // 15 data + zero pad (k==15) + 1 -> coprime with 64 banks
#define TILE_ELEMS      (ROWS_PER_TILE * MTOT)  // 1920 floats = 15 x (32 lanes x float4)
#define TILE_OUT        (ROWS_PER_TILE * 3)     // 384 floats  = 3 x (32 lanes x float4)

// exact f/15 for 0 <= f < 10080 (we only need f < 1920)
__device__ __forceinline__ int div15(int f) { return (f * 8739) >> 17; }

// out[b,s] = sum_k sphc[b,k]^2 * cg[k] * [seg[k]==s]  ==  (x o x) @ W^T  (3x15 GEMM)
// V_WMMA_F32_16X16X4_F32: M=segment, N=batch row (16/group), K chained over 4 chunks of 4.
__global__ __launch_bounds__(256) void l0_contract_wmma(
    const float* __restrict__ sphc,
    const float* __restrict__ cg,
    const int*   __restrict__ seg,
    float* __restrict__ out,
    int nrows)
{
    __shared__ float lds_in [WAVES_PER_BLOCK][ROWS_PER_TILE * LDS_STRIDE];
    __shared__ float lds_out[WAVES_PER_BLOCK][TILE_OUT];

    const int lane   = threadIdx.x & 31;
    const int wid    = threadIdx.x >> 5;
    const int gwave  = blockIdx.x * WAVES_PER_BLOCK + wid;
    const int nwaves = gridDim.x * WAVES_PER_BLOCK;

    float* Lin  = lds_in[wid];
    float* Lout = lds_out[wid];

    // Zero the K==15 padding slots once; data stores never touch k==15.
    // (A's K=15 column is zero, but B must still be finite: 0*Inf/NaN would poison the WMMA sum.)
#pragma unroll
    for (int r = lane; r < ROWS_PER_TILE; r += 32)
        Lin[r * LDS_STRIDE + 15] = 0.0f;

    // A (weights) in the ISA 32-bit A 16x4 layout: lanes 0-15 -> M=lane; VGPR v at half h -> K=2h+v.
    const int m    = lane & 15;
    const int half = lane >> 4;
    v2f Aw[4];
#pragma unroll
    for (int c = 0; c < 4; ++c) {
#pragma unroll
        for (int v = 0; v < 2; ++v) {
            const int k = 4 * c + 2 * half + v;
            float w = 0.0f;
            if (k < MTOT && m < 3) {
                if (seg[k] == m) w = cg[k];
            }
            Aw[c][v] = w;
        }
    }
    __builtin_amdgcn_wave_barrier();

    const int numFullTiles = nrows / ROWS_PER_TILE;
    const int totalElems   = nrows * MTOT;       // <= 60M, fits int32

    for (int tile = gwave; tile < numFullTiles; tile += nwaves) {
        const int base = tile * TILE_ELEMS;

        // Prefetch this wave's next tile into L2 (global_prefetch_b8).
        {
            const int nb = base + nwaves * TILE_ELEMS;
            if (nb < totalElems) __builtin_prefetch(sphc + nb + lane * 32, 0, 1);
        }

        // ---- Load: 15 x global_load_b128 per wave, non-temporal (single-pass stream),
        //      unguarded, fully coalesced & 16B-aligned ----
        v4f tmp[15];
#pragma unroll
        for (int j = 0; j < 15; ++j)
            tmp[j] = __builtin_nontemporal_load((const v4f*)(sphc + base + j * 128 + lane * 4));

        // ---- Scatter to LDS: row-major stride-17 rows; lds_dword = f + 2*(f/15) ----
        // (addresses are tile-invariant -> hoisted out of the loop by the compiler)
#pragma unroll
        for (int j = 0; j < 15; ++j) {
#pragma unroll
            for (int e = 0; e < 4; ++e) {
                const int f = j * 128 + lane * 4 + e;     // flat element index in tile
                Lin[f + 2 * div15(f)] = tmp[j][e];        // == (f/15)*17 + f%15
            }
        }
        __builtin_amdgcn_wave_barrier();

        // ---- Compute: 8 groups of 16 rows, 4 chained WMMAs each (K=0..15, k15 zero pad) ----
        const int n = lane & 15;
#pragma unroll
        for (int gI = 0; gI < 8; ++gI) {
            v8f acc = {};
#pragma unroll
            for (int c = 0; c < 4; ++c) {
                v2f Bv;
#pragma unroll
                for (int v = 0; v < 2; ++v) {
                    const int k = 4 * c + 2 * half + v;   // B layout: K = 2h+v, N = lane&15
                    const float x = Lin[(gI * 16 + n) * LDS_STRIDE + k];
                    Bv[v] = x * x;                        // square fused at B build
                }
                acc = __builtin_amdgcn_wmma_f32_16x16x4_f32(
                        false, Aw[c], false, Bv, (short)0, acc, false, false);
            }
            // Segments s=0..2 live in acc[0..2], lanes 0-15 (D: VGPR r -> M=r, lanes 0-15).
            if (lane < 16) {
#pragma unroll
                for (int s = 0; s < 3; ++s)
                    Lout[(gI * 16 + n) * 3 + s] = acc[s];
            }
        }
        __builtin_amdgcn_wave_barrier();

        // ---- Store: 3 x global_store_b128 per wave, non-temporal, coalesced & 16B-aligned ----
        const int obase = tile * TILE_OUT;
#pragma unroll
        for (int j = 0; j < 3; ++j) {
            const v4f o = *(const v4f*)(Lout + j * 128 + lane * 4);
            __builtin_nontemporal_store(o, (v4f*)(out + obase + j * 128 + lane * 4));
        }
        __builtin_amdgcn_wave_barrier();
    }

    // ---- Tail (nrows % 128 rows): scalar VALU path on wave 0 only; no WMMA, no LDS ----
    if (gwave == 0) {
        const int tailStart = numFullTiles * ROWS_PER_TILE;
        for (int r = tailStart + lane; r < nrows; r += 32) {
            float s0 = 0.f, s1 = 0.f, s2 = 0.f;
#pragma unroll
            for (int k = 0; k < MTOT; ++k) {
                const float x = sphc[r * MTOT + k];
                const float w = x * x * cg[k];
                const int  sg = seg[k];
                s0 += (sg == 0) ? w : 0.f;
                s1 += (sg == 1) ? w : 0.f;
                s2 += (sg == 2) ? w : 0.f;
            }
            out[r * 3 + 0] = s0;
            out[r * 3 + 1] = s1;
            out[r * 3 + 2] = s2;
        }
    }
}

extern "C" void kernel_launch(void* const* d_in, const int* in_sizes, int n_in,
                              void* d_out, int out_size, void* d_ws, size_t ws_size,
                              hipStream_t stream) {
    const float* sphc = (const float*)d_in[0];
    const float* cg   = (const float*)d_in[1];
    const int*   seg  = (const int*)d_in[2];
    float* out = (float*)d_out;

    const int nrows = in_sizes[0] / MTOT;               // B = 4,000,000
    const int numFullTiles = nrows / ROWS_PER_TILE;     // 31,250

    int blocks = (numFullTiles + WAVES_PER_BLOCK - 1) / WAVES_PER_BLOCK;
    if (blocks > 4096) blocks = 4096;                   // grid-stride over tiles
    if (blocks < 1)    blocks = 1;

    l0_contract_wmma<<<blocks, 256, 0, stream>>>(sphc, cg, seg, out, nrows);
}